// SmoothTopKGate_59339268162181
// MI455X (gfx1250) — compile-verified
//
#include <hip/hip_runtime.h>

// SmoothTopKGate for MI455X (gfx1250), wave32.
// One wave handles 16 rows of 64 fp32 values, resident in registers.
// Row-sums of sigma are computed on the matrix pipe via
// V_WMMA_F32_16X16X4_F32 accumulation against an all-ones B matrix.

typedef float v2f __attribute__((ext_vector_type(2)));
typedef float v8f __attribute__((ext_vector_type(8)));

#define ROWS_PER_WAVE 16
#define KCHUNKS 16   // 16 chunks x K=4 = 64 columns
#define N_ITERS 10

__device__ __forceinline__ float fast_tanh(float x) {
#if __has_builtin(__builtin_amdgcn_tanhf)
  return __builtin_amdgcn_tanhf(x);          // v_tanh_f32 (TRANS, gfx1250)
#else
  // tanh(x) = 1 - 2/(exp2(2x*log2e)+1)
  float e = __builtin_amdgcn_exp2f(x * 2.885390082f);
  return 1.0f - 2.0f * __builtin_amdgcn_rcpf(e + 1.0f);
#endif
}

__global__ __launch_bounds__(256)
void smooth_topk_gate_kernel(const float* __restrict__ s,
                             float* __restrict__ g) {
  const int lane  = threadIdx.x & 31;
  const int wave  = threadIdx.x >> 5;
  const long gwave = (long)blockIdx.x * 8 + wave;
  const long rowBase = gwave * ROWS_PER_WAVE;

  // WMMA A-matrix (16x4 f32) layout: lanes 0-15 hold K=0,1 ; lanes 16-31 hold K=2,3
  const int mrow  = lane & 15;   // matrix row owned by this lane (and lane^16)
  const int khalf = lane >> 4;   // which half of each K=4 chunk

  const float* p = s + (rowBase + mrow) * 64 + khalf * 2;

  // ---- load 16 rows x 64 cols into registers (32 VGPRs), scale by 1/(2*tau) = 50
  v2f a[KCHUNKS];
#pragma unroll
  for (int c = 0; c < KCHUNKS; ++c)
    a[c] = *(const v2f*)(p + 4 * c);
#pragma unroll
  for (int c = 0; c < KCHUNKS; ++c) {
    a[c].x *= 50.0f;
    a[c].y *= 50.0f;
  }

  v2f ones;
  ones.x = 1.0f;
  ones.y = 1.0f;

  // per-lane constants for routing row-sums out of the D matrix
  const bool b0 = (lane & 1) != 0;
  const bool b1 = (lane & 2) != 0;
  const bool b2 = (lane & 4) != 0;
  const bool sw = (((lane >> 3) ^ (lane >> 4)) & 1) != 0;   // lanes 8..23 swap halves

  // t = theta * 50 ; sigma = 0.5 + 0.5*tanh(a - t)
  float t = 0.0f;

  for (int it = 0; it < N_ITERS; ++it) {
    v8f acc = {};  // D/C accumulator: 16x16 f32, rowsum replicated across N
#pragma unroll
    for (int c = 0; c < KCHUNKS; ++c) {
      v2f th;
      th.x = fast_tanh(a[c].x - t);
      th.y = fast_tanh(a[c].y - t);
      // acc[m][n] += sum_k th[m][k]  (B == ones)
      acc = __builtin_amdgcn_wmma_f32_16x16x4_f32(
          false, th, false, ones, (short)0, acc, false, false);
    }
    // D layout: vgpr r, lanes 0-15 -> M=r ; lanes 16-31 -> M=r+8 (replicated over N).
    // Select acc[row & 7] per lane, then swap lanes 8..15 <-> 16..23 so each lane
    // ends up with the sum of ITS row (row = lane % 16).
    float s01 = b0 ? acc[1] : acc[0];
    float s23 = b0 ? acc[3] : acc[2];
    float s45 = b0 ? acc[5] : acc[4];
    float s67 = b0 ? acc[7] : acc[6];
    float s03 = b1 ? s23 : s01;
    float s47 = b1 ? s67 : s45;
    float v   = b2 ? s47 : s03;
    float w   = __shfl_xor(v, 24, 32);
    float R   = sw ? w : v;            // R = sum_j tanh(...) over this lane's row
    // sigma_sum = 32 + R/2 ; t += 0.5*(sigma_sum - 8) = 12 + 0.25*R
    t = t + 12.0f + 0.25f * R;
  }

  // ---- final gate with converged theta, streamed straight back out
  float* q = g + (rowBase + mrow) * 64 + khalf * 2;
#pragma unroll
  for (int c = 0; c < KCHUNKS; ++c) {
    v2f o;
    o.x = 0.5f + 0.5f * fast_tanh(a[c].x - t);
    o.y = 0.5f + 0.5f * fast_tanh(a[c].y - t);
    *(v2f*)(q + 4 * c) = o;
  }
}

extern "C" void kernel_launch(void* const* d_in, const int* in_sizes, int n_in,
                              void* d_out, int out_size, void* d_ws, size_t ws_size,
                              hipStream_t stream) {
  const float* s = (const float*)d_in[0];
  float* g = (float*)d_out;
  const int nrows = in_sizes[0] / 64;            // 262144
  const int rows_per_block = 8 * ROWS_PER_WAVE;  // 8 waves * 16 rows = 128
  const int blocks = nrows / rows_per_block;     // 2048 (exact)
  smooth_topk_gate_kernel<<<blocks, 256, 0, stream>>>(s, g);
}